// GFS_84284438217444
// MI455X (gfx1250) — compile-verified
//
#include <hip/hip_runtime.h>
#include <hip/hip_bf16.h>
#include <math.h>

typedef float v2f __attribute__((ext_vector_type(2)));
typedef float v8f __attribute__((ext_vector_type(8)));

#define EPSC 1e-5f
#define LOG_AVG 3.4965075614664802f   // log(32 + 1)

__device__ __forceinline__ unsigned fenc(float f) {
    unsigned u = __float_as_uint(f);
    return (u >> 31) ? ~u : (u | 0x80000000u);
}
__device__ __forceinline__ float fdec(unsigned u) {
    return (u & 0x80000000u) ? __uint_as_float(u ^ 0x80000000u)
                             : __uint_as_float(~u);
}

// ---- K0: init per-dst accumulators (ws is poisoned; must re-init every call)
__global__ void k_init(float* deg, float* sum, float* sum2,
                       unsigned* mn, unsigned* mx, int Nd) {
    long long tid = (long long)blockIdx.x * 256 + threadIdx.x;
    long long n = (long long)Nd * 32;
    if (tid < n) {
        sum[tid]  = 0.f;
        sum2[tid] = 0.f;
        mn[tid]   = 0xFFFFFFFFu;  // encodes +inf
        mx[tid]   = 0u;           // encodes -inf
    }
    if (tid < Nd) deg[tid] = 0.f;
}

// ---- K0b: permute W_agg/W1/W2 into per-lane WMMA-B fragment order (v2f)
// pAgg[((k*48+c)*32+l)*2+i] = W_agg[(k*192 + c*4 + (l>>4)*2 + i)*16 + (l&15)]
__global__ void k_perm(const float* __restrict__ W_agg,
                       const float* __restrict__ W1,
                       const float* __restrict__ W2,
                       float* __restrict__ pAgg,
                       float* __restrict__ pW1,
                       float* __restrict__ pW2) {
    int tid = blockIdx.x * 256 + threadIdx.x;
    if (tid < 2 * 48 * 64) {
        int i = tid & 1, l = (tid >> 1) & 31, c = (tid >> 6) % 48, k = tid / (64 * 48);
        int f = c * 4 + (l >> 4) * 2 + i, n = l & 15;
        pAgg[tid] = W_agg[(k * 192 + f) * 16 + n];
    }
    int t1 = tid - 2 * 48 * 64;
    if (t1 >= 0 && t1 < 4 * 24 * 64) {
        int i = t1 & 1, l = (t1 >> 1) & 31, c = (t1 >> 6) % 24, nt = t1 / (64 * 24);
        int f = c * 4 + (l >> 4) * 2 + i, n = nt * 16 + (l & 15);
        pW1[t1] = W1[f * 64 + n];
    }
    int t2 = t1 - 4 * 24 * 64;
    if (t2 >= 0 && t2 < 2 * 16 * 64) {
        int i = t2 & 1, l = (t2 >> 1) & 31, c = (t2 >> 6) % 16, nt = t2 / (64 * 16);
        int f = c * 4 + (l >> 4) * 2 + i, n = nt * 16 + (l & 15);
        pW2[t2] = W2[f * 32 + n];
    }
}

// ---- K1: h_src[n][k][d]  (k=0 mean of 4 field embeddings, k=1 max)
__global__ void k_hsrc(const int* __restrict__ Cat_src,
                       const float* __restrict__ w,
                       float* __restrict__ h, long long Ns) {
    long long tid = (long long)blockIdx.x * 256 + threadIdx.x;
    if (tid >= Ns * 16) return;
    long long n = tid >> 4;
    int d = (int)(tid & 15);
    float acc = 0.f, mxv = -INFINITY;
    #pragma unroll
    for (int f = 0; f < 4; ++f) {
        int idx = Cat_src[n * 4 + f];
        float v = w[(long long)idx * 16 + d];
        acc += v;
        mxv = fmaxf(mxv, v);
    }
    h[n * 32 + d]      = acc * 0.25f;
    h[n * 32 + 16 + d] = mxv;
}

// ---- K2: edge scatter. e is wave-uniform -> scalar id loads; h read coalesced.
// Accumulators are L2-resident (51 MB << 192 MB L2).
__global__ void k_edge(const int* __restrict__ src_ids,
                       const int* __restrict__ dst_ids,
                       const float* __restrict__ h,
                       float* deg, float* sum, float* sum2,
                       unsigned* mn, unsigned* mx, long long E) {
    int wave = threadIdx.x >> 5;
    int lane = threadIdx.x & 31;
    long long e = (long long)blockIdx.x * 8 + wave;   // wave-uniform
    if (e >= E) return;
    int s  = src_ids[e];
    int dd = dst_ids[e];
    float m = h[(long long)s * 32 + lane];
    long long idx = (long long)dd * 32 + lane;
    atomicAdd(&sum[idx],  m);
    atomicAdd(&sum2[idx], m * m);
    atomicMin(&mn[idx], fenc(m));
    atomicMax(&mx[idx], fenc(m));
    if (lane == 0) atomicAdd(&deg[dd], 1.f);
}

// ---- WMMA fragment chain: NC chunks of K=4, A from LDS (v2f, scaled), B from
// pre-permuted global table (v2f, coalesced, L2-resident).
template <int NC>
__device__ __forceinline__ v8f gemm_frag(const float* __restrict__ Arow, float sc,
                                         const v2f* __restrict__ Bp, v8f acc,
                                         int half) {
    #pragma unroll
    for (int c = 0; c < NC; ++c) {
        v2f a = *(const v2f*)(Arow + c * 4 + half * 2);
        a.x *= sc; a.y *= sc;
        v2f b = Bp[c * 32];
        acc = __builtin_amdgcn_wmma_f32_16x16x4_f32(
                  false, a, false, b, (short)0, acc, false, false);
    }
    return acc;
}

// ---- K3: per-dst: stats -> PNA scalers -> WMMA einsum -> DeepFM (WMMA MLP)
// One wave handles 16 dst nodes; 2 waves per block (37 KB LDS).
#define WAVE_LDS 4640   // tmp 2048 | flat 1536 | h1 1024 | amp 16 | att 16
__global__ void __launch_bounds__(64)
k_dst(const int* __restrict__ Cat_dst, const float* __restrict__ w,
      const float* __restrict__ deg, const float* __restrict__ sum,
      const float* __restrict__ sum2,
      const unsigned* __restrict__ mnk, const unsigned* __restrict__ mxk,
      const float* __restrict__ pAgg, const float* __restrict__ b_agg,
      const float* __restrict__ w_lin, const float* __restrict__ b_lin,
      const float* __restrict__ pW1, const float* __restrict__ b1,
      const float* __restrict__ pW2, const float* __restrict__ b2,
      const float* __restrict__ W3, const float* __restrict__ b3,
      float* __restrict__ out, int Nd) {
    __shared__ float lds[2 * WAVE_LDS];
    int lane = threadIdx.x & 31;
    int wave = threadIdx.x >> 5;
    float* T   = lds + wave * WAVE_LDS;  // tmp[16 nodes][k=2][64 stats]
    float* FLT = T + 2048;               // flat[16][96]
    float* H1  = T + 3584;               // h1[16][64]
    float* AMP = T + 4608;
    float* ATT = T + 4624;
    float* H2  = T;                      // overlay tmp after it's consumed

    int tile  = blockIdx.x * 2 + wave;
    int node0 = tile * 16;

    // Phase A: aggregate stats -> tmp (lane = (k,d) slot -> coalesced reads)
    for (int t = 0; t < 16; ++t) {           // t = node within tile
        int node = min(node0 + t, Nd - 1);
        int k = lane >> 4, d = lane & 15;
        long long idx = (long long)node * 32 + lane;
        float dv   = deg[node];
        float safe = fmaxf(dv, 1.f);
        float mean  = sum[idx] / safe;
        float mean2 = sum2[idx] / safe;
        bool  has   = dv > 0.f;
        float mnv = has ? fdec(mnk[idx]) : 0.f;
        float mxv = has ? fdec(mxk[idx]) : 0.f;
        float sd  = sqrtf(fmaxf(mean2 - mean * mean, 0.f) + EPSC);
        int base = ((t * 2 + k) << 6) + d;   // [mean|mn|mx|std] blocks of 16
        T[base]      = mean;
        T[base + 16] = mnv;
        T[base + 32] = mxv;
        T[base + 48] = sd;
    }
    if (lane < 16) {
        int node = min(node0 + lane, Nd - 1);
        float logd = logf(deg[node] + 1.f);
        AMP[lane] = logd / LOG_AVG;
        ATT[lane] = (logd > 0.f) ? (LOG_AVG / fmaxf(logd, EPSC)) : 0.f;
    }
    // dst embeddings -> flat[:,0:64]; lane innermost -> coalesced w reads
    for (int t = 0; t < 32; ++t) {
        int g = t * 32 + lane;               // 0..1023 = 16 nodes * 64
        int m = g >> 6, rem = g & 63;
        int f = rem >> 4, d = rem & 15;
        int node = min(node0 + m, Nd - 1);
        int cid = Cat_dst[node * 4 + f];
        FLT[m * 96 + f * 16 + d] = w[(long long)cid * 16 + d];
    }
    __syncthreads();

    int half = lane >> 4;   // K-half for A/B fragments
    int nn   = lane & 15;   // row (A) / col (B,C)
    float ampv = AMP[nn], attv = ATT[nn];
    const v2f* pAggV = (const v2f*)pAgg;
    const v2f* pW1V  = (const v2f*)pW1;
    const v2f* pW2V  = (const v2f*)pW2;

    // Phase B: mes[k] = scaled(16x192) @ W_agg[k](192x16) + b_agg[k]
    // Chunks 0-15: identity; 16-31: *amp; 32-47: *att (same 64 raw stats).
    #pragma unroll
    for (int k = 0; k < 2; ++k) {
        const float* Arow = T + (((nn << 1) + k) << 6);
        const v2f* Bp = pAggV + k * 48 * 32 + lane;
        v8f acc = {};
        acc = gemm_frag<16>(Arow, 1.f,  Bp,           acc, half);
        acc = gemm_frag<16>(Arow, ampv, Bp + 16 * 32, acc, half);
        acc = gemm_frag<16>(Arow, attv, Bp + 32 * 32, acc, half);
        float bias = b_agg[k * 16 + nn];
        #pragma unroll
        for (int r = 0; r < 8; ++r) {
            int m = r + half * 8;
            FLT[m * 96 + 64 + k * 16 + nn] = acc[r] + bias;
        }
    }
    __syncthreads();

    // Phase C: linear + FM terms (lanes 0..15, one node each; scalar VALU)
    float linfm = 0.f;
    if (lane < 16) {
        int m = lane;
        float lin = b_lin[0];
        for (int f = 0; f < 96; ++f) lin += FLT[m * 96 + f] * w_lin[f];
        float fm = 0.f;
        for (int d = 0; d < 16; ++d) {
            float s = 0.f, ss = 0.f;
            #pragma unroll
            for (int f = 0; f < 6; ++f) {
                float v = FLT[m * 96 + f * 16 + d];
                s += v; ss += v * v;
            }
            fm += s * s - ss;
        }
        linfm = lin + 0.5f * fm;
    }

    // Phase D1: h1 = relu(flat(16x96) @ W1(96x64) + b1)
    #pragma unroll
    for (int nt = 0; nt < 4; ++nt) {
        v8f acc = {};
        acc = gemm_frag<24>(FLT + nn * 96, 1.f, pW1V + nt * 24 * 32 + lane, acc, half);
        float bias = b1[nt * 16 + nn];
        #pragma unroll
        for (int r = 0; r < 8; ++r) {
            int m = r + half * 8;
            H1[m * 64 + nt * 16 + nn] = fmaxf(acc[r] + bias, 0.f);
        }
    }
    __syncthreads();

    // Phase D2: h2 = relu(h1(16x64) @ W2(64x32) + b2), stored over tmp region
    #pragma unroll
    for (int nt = 0; nt < 2; ++nt) {
        v8f acc = {};
        acc = gemm_frag<16>(H1 + nn * 64, 1.f, pW2V + nt * 16 * 32 + lane, acc, half);
        float bias = b2[nt * 16 + nn];
        #pragma unroll
        for (int r = 0; r < 8; ++r) {
            int m = r + half * 8;
            H2[m * 32 + nt * 16 + nn] = fmaxf(acc[r] + bias, 0.f);
        }
    }
    __syncthreads();

    // Final: deep head + sigmoid
    if (lane < 16) {
        int m = lane;
        int node = node0 + m;
        float dp = b3[0];
        for (int j = 0; j < 32; ++j) dp += H2[m * 32 + j] * W3[j];
        float x = linfm + dp;
        if (node < Nd) out[node] = 1.f / (1.f + expf(-x));
    }
}

// ---- K4: second tuple output (label), written after predictions
__global__ void k_label(const int* __restrict__ label, float* __restrict__ out, int Nd) {
    int tid = blockIdx.x * 256 + threadIdx.x;
    if (tid < Nd) out[Nd + tid] = (float)label[tid];
}

extern "C" void kernel_launch(void* const* d_in, const int* in_sizes, int n_in,
                              void* d_out, int out_size, void* d_ws, size_t ws_size,
                              hipStream_t stream) {
    const int*   Cat_src = (const int*)d_in[0];
    const int*   Cat_dst = (const int*)d_in[1];
    const int*   src_ids = (const int*)d_in[2];
    const int*   dst_ids = (const int*)d_in[3];
    const int*   label   = (const int*)d_in[4];
    const float* w       = (const float*)d_in[5];
    const float* W_agg   = (const float*)d_in[6];
    const float* b_agg   = (const float*)d_in[7];
    const float* w_lin   = (const float*)d_in[8];
    const float* b_lin   = (const float*)d_in[9];
    const float* W1      = (const float*)d_in[10];
    const float* b1      = (const float*)d_in[11];
    const float* W2      = (const float*)d_in[12];
    const float* b2      = (const float*)d_in[13];
    const float* W3      = (const float*)d_in[14];
    const float* b3      = (const float*)d_in[15];

    long long Ns = (long long)in_sizes[0] / 4;  // Fs = 4
    int       Nd = in_sizes[1] / 4;             // Fd = 4
    long long E  = (long long)in_sizes[2];

    // Workspace: h_src | deg | sum | sum2 | mnkey | mxkey | pAgg | pW1 | pW2
    char* ws = (char*)d_ws;
    float* h_src = (float*)ws;    ws += (size_t)Ns * 32 * sizeof(float);
    float* deg   = (float*)ws;    ws += (size_t)Nd * sizeof(float);
    float* sum   = (float*)ws;    ws += (size_t)Nd * 32 * sizeof(float);
    float* sum2  = (float*)ws;    ws += (size_t)Nd * 32 * sizeof(float);
    unsigned* mn = (unsigned*)ws; ws += (size_t)Nd * 32 * sizeof(unsigned);
    unsigned* mx = (unsigned*)ws; ws += (size_t)Nd * 32 * sizeof(unsigned);
    float* pAgg  = (float*)ws;    ws += (size_t)2 * 48 * 64 * sizeof(float);
    float* pW1   = (float*)ws;    ws += (size_t)4 * 24 * 64 * sizeof(float);
    float* pW2   = (float*)ws;

    long long nInit = (long long)Nd * 32;
    k_init<<<(unsigned)((nInit + 255) / 256), 256, 0, stream>>>(deg, sum, sum2, mn, mx, Nd);
    k_perm<<<(2 * 48 * 64 + 4 * 24 * 64 + 2 * 16 * 64 + 255) / 256, 256, 0, stream>>>(
        W_agg, W1, W2, pAgg, pW1, pW2);
    k_hsrc<<<(unsigned)((Ns * 16 + 255) / 256), 256, 0, stream>>>(Cat_src, w, h_src, Ns);
    k_edge<<<(unsigned)((E + 7) / 8), 256, 0, stream>>>(
        src_ids, dst_ids, h_src, deg, sum, sum2, mn, mx, E);
    int tiles = (Nd + 15) / 16;
    k_dst<<<(unsigned)((tiles + 1) / 2), 64, 0, stream>>>(
        Cat_dst, w, deg, sum, sum2, mn, mx, pAgg, b_agg,
        w_lin, b_lin, pW1, b1, pW2, b2, W3, b3, (float*)d_out, Nd);
    k_label<<<(unsigned)((Nd + 255) / 256), 256, 0, stream>>>(label, (float*)d_out, Nd);
}